// Conv1DBlock_49357764166239
// MI455X (gfx1250) — compile-verified
//
#include <hip/hip_runtime.h>

// CDNA5 / gfx1250 fused Conv1D block (Conv-TasNet style) in fp32 WMMA.
//   h   = prelu(x @ Wb, alpha1)                      (GEMM1, kept in LDS)
//   y   = prelu(depthwise3_dil4(h), alpha2)          (computed on the fly)
//   res = y @ Wr + x ;  skip = y @ Ws                (GEMM2/3, shared A-frags)

typedef __attribute__((ext_vector_type(2))) float v2f;
typedef __attribute__((ext_vector_type(8))) float v8f;

#define B_      8
#define T_      4096
#define C_      128          // bottleneck channels (Bc)
#define H_      512          // hidden channels
#define DIL     4
#define TILE_T  64           // output timesteps per workgroup
#define HROWS   80           // 64 + 2*4 halo, padded to 5 * 16 for WMMA M-tiling
#define NMT     (HROWS / 16) // 5 M-tiles in phase 1
#define LDST    (H_ + 4)     // LDS row stride (floats), pad kills bank conflicts

__global__ __launch_bounds__(256)
void conv1d_block_fused(const float* __restrict__ x,
                        const float* __restrict__ wb,   // (128, 512)
                        const float* __restrict__ a1,   // (4096, 512)
                        const float* __restrict__ wd,   // (3, 1, 512)
                        const float* __restrict__ a2,   // (4096, 512)
                        const float* __restrict__ wr,   // (512, 128)
                        const float* __restrict__ wsk,  // (512, 128)
                        float* __restrict__ out_res,    // (8, 4096, 128)
                        float* __restrict__ out_skip)   // (8, 4096, 128)
{
    __shared__ float hs[HROWS * LDST];   // ~161 KB: h tile with halo
    __shared__ float wdep[3 * H_];       // 6 KB: depthwise weights

    const int wave = threadIdx.x >> 5;   // 0..7
    const int lane = threadIdx.x & 31;
    const int l16  = lane & 15;
    const int lhi  = lane >> 4;          // 0 or 1

    const int blk  = blockIdx.x;         // 0..511
    const int b    = blk >> 6;           // batch
    const int t0   = (blk & 63) * TILE_T;

    // L2 prefetch hints for the shared, reused weight matrices
    __builtin_prefetch(wb,  0, 1);
    __builtin_prefetch(wr,  0, 1);
    __builtin_prefetch(wsk, 0, 1);

    // stage depthwise weights
    for (int i = threadIdx.x; i < 3 * H_; i += 256) wdep[i] = wd[i];

    const v8f vzero = {0.f,0.f,0.f,0.f,0.f,0.f,0.f,0.f};

    // ================= Phase 1: hs = prelu(x_halo @ Wb, alpha1) ================
    // Wave w owns N-tiles 4w..4w+3 (16 cols each), processed as 2 passes of 2.
    // Per pass: kk-outer / mt-inner so each B fragment feeds 5 M-tiles (10 WMMAs
    // per 4 B-loads + 5 A-b64-loads).

    // Per-mt A-fragment row state (A 16x4: M = lane%16, K = (lane/16)*2 + v)
    const float* xrow[NMT];
    bool tin[NMT];
    #pragma unroll
    for (int mt = 0; mt < NMT; ++mt) {
        const int ta = t0 - DIL + mt * 16 + l16;
        tin[mt]  = (ta >= 0) && (ta < T_);
        xrow[mt] = x + ((size_t)b * T_ + (tin[mt] ? ta : 0)) * C_;
    }

    for (int p = 0; p < 2; ++p) {
        const int n0 = (wave * 4 + p * 2) * 16 + l16;   // first of 2 N-tiles

        v8f acc[NMT][2];
        #pragma unroll
        for (int mt = 0; mt < NMT; ++mt) { acc[mt][0] = vzero; acc[mt][1] = vzero; }

        // bounded unroll: keeps VGPR usage < 256 (no s_set_vgpr_msb churn,
        // ~5 waves/SIMD) while leaving 8 iterations for load clustering.
        #pragma unroll 8
        for (int kk = 0; kk < C_ / 4; ++kk) {            // K = 128, step 4
            const int ka = kk * 4 + lhi * 2;
            v2f bf0, bf1;
            bf0.x = wb[(size_t)(ka    ) * H_ + n0];
            bf0.y = wb[(size_t)(ka + 1) * H_ + n0];
            bf1.x = wb[(size_t)(ka    ) * H_ + n0 + 16];
            bf1.y = wb[(size_t)(ka + 1) * H_ + n0 + 16];
            #pragma unroll
            for (int mt = 0; mt < NMT; ++mt) {
                const float2 av = *(const float2*)(xrow[mt] + ka);  // b64, in-bounds
                v2f a;
                a.x = tin[mt] ? av.x : 0.f;                         // cndmask, no branch
                a.y = tin[mt] ? av.y : 0.f;
                acc[mt][0] = __builtin_amdgcn_wmma_f32_16x16x4_f32(
                                 false, a, false, bf0, (short)0, acc[mt][0], false, false);
                acc[mt][1] = __builtin_amdgcn_wmma_f32_16x16x4_f32(
                                 false, a, false, bf1, (short)0, acc[mt][1], false, false);
            }
        }

        // PReLU(alpha1) + store tiles to LDS
        // C/D 16x16: VGPR i -> M = i (lanes 0-15) / 8+i (lanes 16-31), N = lane%16
        #pragma unroll
        for (int mt = 0; mt < NMT; ++mt) {
            #pragma unroll
            for (int j = 0; j < 2; ++j) {
                const int n = n0 + j * 16;
                #pragma unroll
                for (int i = 0; i < 8; ++i) {
                    const int r = mt * 16 + lhi * 8 + i;
                    int tt = t0 - DIL + r;
                    tt = tt < 0 ? 0 : (tt > T_ - 1 ? T_ - 1 : tt);  // clamp for alpha read
                    const float al = a1[(size_t)tt * H_ + n];
                    float v = acc[mt][j][i];
                    v = fmaxf(v, 0.f) + al * fminf(v, 0.f);
                    hs[r * LDST + n] = v;
                }
            }
        }
    }
    __syncthreads();

    // ==== Phase 2: y = prelu(conv(hs), alpha2); res = y@Wr + x; skip = y@Ws ====
    // 4 M-tiles x 8 N-tiles, two waves per M-tile, 4 N-tiles per wave.
    // A fragment (y) computed on the fly from LDS and shared by res & skip chains.
    const int mt2  = wave >> 1;              // M-tile 0..3
    const int ntb  = (wave & 1) * 4;         // N-tile base 0 or 4
    const int ry   = mt2 * 16 + l16;         // local output row 0..63 (A row)
    const int tabs = t0 + ry;                // absolute timestep (always in range)

    const float* h0p = &hs[(ry          ) * LDST];
    const float* h1p = &hs[(ry + DIL    ) * LDST];
    const float* h2p = &hs[(ry + 2 * DIL) * LDST];
    const float* a2p = a2 + (size_t)tabs * H_;

    v8f cres[4], cskp[4];
    #pragma unroll
    for (int j = 0; j < 4; ++j) { cres[j] = vzero; cskp[j] = vzero; }

    #pragma unroll 4
    for (int kk = 0; kk < H_ / 4; ++kk) {    // K = 512, step 4
        const int k0 = kk * 4 + lhi * 2;
        // dilated depthwise conv + PReLU -> A fragment (all b64 accesses)
        const float2 h0 = *(const float2*)(h0p + k0);
        const float2 h1 = *(const float2*)(h1p + k0);
        const float2 h2 = *(const float2*)(h2p + k0);
        const float2 w0 = *(const float2*)(wdep + k0);
        const float2 w1 = *(const float2*)(wdep + H_ + k0);
        const float2 w2 = *(const float2*)(wdep + 2 * H_ + k0);
        const float2 al = *(const float2*)(a2p + k0);
        v2f a;
        {
            const float yx = w0.x * h0.x + w1.x * h1.x + w2.x * h2.x;
            const float yy = w0.y * h0.y + w1.y * h1.y + w2.y * h2.y;
            a.x = fmaxf(yx, 0.f) + al.x * fminf(yx, 0.f);
            a.y = fmaxf(yy, 0.f) + al.y * fminf(yy, 0.f);
        }
        #pragma unroll
        for (int j = 0; j < 4; ++j) {
            const int n = (ntb + j) * 16 + l16;
            v2f br, bs;
            br.x = wr [(size_t)(k0    ) * C_ + n];
            br.y = wr [(size_t)(k0 + 1) * C_ + n];
            bs.x = wsk[(size_t)(k0    ) * C_ + n];
            bs.y = wsk[(size_t)(k0 + 1) * C_ + n];
            cres[j] = __builtin_amdgcn_wmma_f32_16x16x4_f32(
                          false, a, false, br, (short)0, cres[j], false, false);
            cskp[j] = __builtin_amdgcn_wmma_f32_16x16x4_f32(
                          false, a, false, bs, (short)0, cskp[j], false, false);
        }
    }

    // epilogue: residual add + stores
    #pragma unroll
    for (int j = 0; j < 4; ++j) {
        const int n = (ntb + j) * 16 + l16;
        #pragma unroll
        for (int i = 0; i < 8; ++i) {
            const int r = mt2 * 16 + lhi * 8 + i;           // output row 0..63
            const size_t idx = ((size_t)b * T_ + (t0 + r)) * C_ + n;
            out_res[idx]  = cres[j][i] + x[idx];
            out_skip[idx] = cskp[j][i];
        }
    }
}

extern "C" void kernel_launch(void* const* d_in, const int* in_sizes, int n_in,
                              void* d_out, int out_size, void* d_ws, size_t ws_size,
                              hipStream_t stream) {
    const float* x   = (const float*)d_in[0];
    const float* wb  = (const float*)d_in[1];
    const float* a1  = (const float*)d_in[2];
    const float* wd  = (const float*)d_in[3];
    const float* a2  = (const float*)d_in[4];
    const float* wr  = (const float*)d_in[5];
    const float* wsk = (const float*)d_in[6];

    float* res = (float*)d_out;
    float* skp = res + (size_t)B_ * T_ * C_;

    dim3 grid(B_ * (T_ / TILE_T));   // 512 workgroups
    dim3 block(256);                 // 8 wave32s
    hipLaunchKernelGGL(conv1d_block_fused, grid, block, 0, stream,
                       x, wb, a1, wd, a2, wr, wsk, res, skp);
}